// MahalanobisDistance_20263655703036
// MI455X (gfx1250) — compile-verified
//
#include <hip/hip_runtime.h>
#include <math.h>

typedef float v2f __attribute__((ext_vector_type(2)));
typedef float v8f __attribute__((ext_vector_type(8)));

#define BB 16
#define NN 3136
#define DD 256
#define LDSW 260   // 256 + 4 pad: row stride 260 -> lane m hits bank (m*4)%64, conflict-free

__global__ __launch_bounds__(256)
void mahalanobis_wmma_f32_kernel(const float* __restrict__ features,
                                 const float* __restrict__ mean,
                                 const float* __restrict__ cov_inv,
                                 float* __restrict__ out)
{
    __shared__ float lds_delta[BB * LDSW];   // delta, padded rows
    __shared__ float lds_mean[DD];
    __shared__ float lds_partial[8][BB];

    const int n    = blockIdx.x;             // one workgroup per n
    const int tid  = threadIdx.x;            // 0..255 (8 waves)
    const int lane = tid & 31;
    const int wave = tid >> 5;               // 0..7
    const int hh   = (lane >> 4) & 1;        // lane half (wave32)
    const int lm   = lane & 15;

    // ---- stage mean[n][:] ----
    lds_mean[tid] = mean[(size_t)n * DD + tid];
    __syncthreads();

    // ---- delta = features[b][n][:] - mean[n][:]  -> LDS (coalesced float4 loads) ----
    {
        const int b  = tid >> 4;             // 16 threads per b-row
        const int d0 = (tid & 15) * 16;      // 16 consecutive d per thread
        const float* frow = features + ((size_t)b * NN + n) * DD + d0;
        float* drow = lds_delta + b * LDSW + d0;
        #pragma unroll
        for (int i = 0; i < 16; i += 4) {
            float4 f = *(const float4*)(frow + i);
            drow[i + 0] = f.x - lds_mean[d0 + i + 0];
            drow[i + 1] = f.y - lds_mean[d0 + i + 1];
            drow[i + 2] = f.z - lds_mean[d0 + i + 2];
            drow[i + 3] = f.w - lds_mean[d0 + i + 3];
        }
    }
    __syncthreads();

    const float* cov = cov_inv + (size_t)n * DD * DD;   // row-major (d, e), symmetric

    // Each wave handles output column tiles j0 = wave and j1 = wave + 8 (16 cols each).
    const int col0 = wave * 16 + lm;
    const int col1 = (wave + 8) * 16 + lm;

    v8f acc0 = {};
    v8f acc1 = {};

    // A-fragment base: lane holds row M = lm; VGPR pair = K {k+2*hh, k+2*hh+1}
    const float* arow = lds_delta + lm * LDSW + 2 * hh;

    #pragma unroll 4
    for (int k = 0; k < DD; k += 4) {
        // A: 16x4 tile of delta, conflict-free ds_load_b64
        v2f a = *(const v2f*)(arow + k);

        // B: 4x16 tile of cov_inv. b.x = B[k+2h][col], b.y = B[k+2h+1][col]
        const int r = k + 2 * hh;
        v2f b0, b1;
        b0.x = cov[(size_t)r       * DD + col0];
        b0.y = cov[(size_t)(r + 1) * DD + col0];
        b1.x = cov[(size_t)r       * DD + col1];
        b1.y = cov[(size_t)(r + 1) * DD + col1];

        acc0 = __builtin_amdgcn_wmma_f32_16x16x4_f32(false, a, false, b0,
                                                     (short)0, acc0, false, false);
        acc1 = __builtin_amdgcn_wmma_f32_16x16x4_f32(false, a, false, b1,
                                                     (short)0, acc1, false, false);
    }

    // ---- epilogue: dist2[b] = sum_e M[b][e] * delta[b][e] ----
    // C/D layout: reg r, lane -> row b = r + 8*hh, col e = tile*16 + lm
    #pragma unroll
    for (int r = 0; r < 8; ++r) {
        const int b = r + 8 * hh;
        const float d0 = lds_delta[b * LDSW + col0];
        const float d1 = lds_delta[b * LDSW + col1];
        float v = acc0[r] * d0 + acc1[r] * d1;
        // reduce across the 16 lanes of this half (the e dimension)
        v += __shfl_xor(v, 1, 32);
        v += __shfl_xor(v, 2, 32);
        v += __shfl_xor(v, 4, 32);
        v += __shfl_xor(v, 8, 32);
        if (lm == 0) lds_partial[wave][b] = v;
    }
    __syncthreads();

    if (tid < BB) {
        float s = 0.0f;
        #pragma unroll
        for (int w = 0; w < 8; ++w) s += lds_partial[w][tid];
        out[(size_t)tid * NN + n] = sqrtf(fmaxf(s, 0.0f));
    }
}

extern "C" void kernel_launch(void* const* d_in, const int* in_sizes, int n_in,
                              void* d_out, int out_size, void* d_ws, size_t ws_size,
                              hipStream_t stream) {
    const float* features = (const float*)d_in[0];   // (B, N, D) f32
    const float* mean     = (const float*)d_in[1];   // (N, D)    f32
    const float* cov_inv  = (const float*)d_in[2];   // (N, D, D) f32
    float* out = (float*)d_out;                      // (B, N) -> reshaped (B, 56, 56)

    mahalanobis_wmma_f32_kernel<<<NN, 256, 0, stream>>>(features, mean, cov_inv, out);
}